// GRUMappingNoGCN_47373489274957
// MI455X (gfx1250) — compile-verified
//
#include <hip/hip_runtime.h>
#include <math.h>

typedef float v2f __attribute__((ext_vector_type(2)));
typedef float v8f __attribute__((ext_vector_type(8)));

constexpr int kB  = 2;
constexpr int kN1 = 4096;
constexpr int kN2 = 8192;
constexpr int kC  = 64;
constexpr int kK  = 16;   // nsample
#define LEAKK 0.1f

// ---------------------------------------------------------------------------
// fp32 WMMA wrapper: D = A(16x4) * B(4x16) + C(16x16)
// ---------------------------------------------------------------------------
__device__ __forceinline__ v8f wmma4(v2f a, v2f b, v8f c) {
  return __builtin_amdgcn_wmma_f32_16x16x4_f32(false, a, false, b, (short)0, c,
                                               false, false);
}

__device__ __forceinline__ float leaky_(float x) { return x > 0.0f ? x : LEAKK * x; }
__device__ __forceinline__ float sigmoid_(float x) { return 1.0f / (1.0f + expf(-x)); }

// Async global->LDS copy, 16B per lane, per-instruction immediate offset is
// added to BOTH the LDS and the global address (CDNA5 ISA).
#define ASYNC_LDS_B128(ldsaddr, gaddr, OFF)                                    \
  asm volatile("global_load_async_to_lds_b128 %0, %1, off offset:" #OFF        \
               :: "v"(ldsaddr), "v"(gaddr) : "memory")

// Copy one 16x64-float f2n tile (per-lane: 128 contiguous bytes = 8 x b128).
__device__ __forceinline__ void issue_ftile_async(unsigned ldsaddr, unsigned long long g) {
  ASYNC_LDS_B128(ldsaddr, g, 0);
  ASYNC_LDS_B128(ldsaddr, g, 16);
  ASYNC_LDS_B128(ldsaddr, g, 32);
  ASYNC_LDS_B128(ldsaddr, g, 48);
  ASYNC_LDS_B128(ldsaddr, g, 64);
  ASYNC_LDS_B128(ldsaddr, g, 80);
  ASYNC_LDS_B128(ldsaddr, g, 96);
  ASYNC_LDS_B128(ldsaddr, g, 112);
}

// ---------------------------------------------------------------------------
// Kernel 1: transpose + normalize features; pad xyz; |x2|^2
//   which==0 -> N1 set (f1n, x1t), which==1 -> N2 set (f2n, x2t, n2sq)
// ---------------------------------------------------------------------------
__global__ void prep_kernel(const float* __restrict__ xyz1,
                            const float* __restrict__ xyz2,
                            const float* __restrict__ knn1,
                            const float* __restrict__ knn2,
                            float* __restrict__ f1n, float* __restrict__ f2n,
                            float* __restrict__ x1t, float* __restrict__ x2t,
                            float* __restrict__ n2sq) {
  const int b = blockIdx.z;
  const int which = blockIdx.y;
  const int n = blockIdx.x * blockDim.x + threadIdx.x;
  const int N = which ? kN2 : kN1;
  if (n >= N) return;

  const float* knn = which ? knn2 : knn1;
  const float* xyz = which ? xyz2 : xyz1;
  float* f  = which ? f2n : f1n;
  float* xt = which ? x2t : x1t;

  const float* kb = knn + (size_t)b * kC * N + n;
  float ss = 0.0f;
  #pragma unroll 8
  for (int c = 0; c < kC; ++c) {
    float v = kb[(size_t)c * N];
    ss += v * v;
  }
  const float rinv = rsqrtf(ss + 1e-8f);
  float* fo = f + ((size_t)b * N + n) * kC;
  #pragma unroll 8
  for (int c = 0; c < kC; ++c) fo[c] = kb[(size_t)c * N] * rinv;

  const float* xb = xyz + (size_t)b * 3 * N + n;
  const float x = xb[0], y = xb[(size_t)N], z = xb[2 * (size_t)N];
  float* xo = xt + ((size_t)b * N + n) * 4;
  xo[0] = x; xo[1] = y; xo[2] = z; xo[3] = 0.0f;
  if (which) n2sq[(size_t)b * N + n] = x * x + y * y + z * z;
}

// ---------------------------------------------------------------------------
// Kernel 2: fused 1x1 convs  out[b][n][o] = sum_c W[o][c] * points[b][c][n]
// One wave per 16-point tile; 4 output tiles of 16 chans; fp32 WMMA chain.
// ---------------------------------------------------------------------------
__global__ __launch_bounds__(32) void fuse_kernel(
    const float* __restrict__ points1, const float* __restrict__ points2,
    const float* __restrict__ Wfr, const float* __restrict__ Wfro,
    const float* __restrict__ Wfz, const float* __restrict__ Wfr2,
    const float* __restrict__ Wfro2, const float* __restrict__ Wfz2,
    float* __restrict__ gp1r, float* __restrict__ p1o,
    float* __restrict__ gp1z, float* __restrict__ p2r,
    float* __restrict__ p2ro, float* __restrict__ p2z) {
  const int combo = blockIdx.y;
  const int b = blockIdx.z;
  const float* src; const float* W; float* dst; int N;
  switch (combo) {
    case 0:  src = points1; W = Wfr;   dst = gp1r; N = kN1; break;
    case 1:  src = points1; W = Wfro;  dst = p1o;  N = kN1; break;
    case 2:  src = points1; W = Wfz;   dst = gp1z; N = kN1; break;
    case 3:  src = points2; W = Wfr2;  dst = p2r;  N = kN2; break;
    case 4:  src = points2; W = Wfro2; dst = p2ro; N = kN2; break;
    default: src = points2; W = Wfz2;  dst = p2z;  N = kN2; break;
  }
  const int nbase = blockIdx.x * 16;
  if (nbase >= N) return;

  const int l = threadIdx.x, lm = l & 15, lh = l >> 4;
  const float* sb = src + (size_t)b * kC * N;

  v8f acc[4] = {};
  #pragma unroll
  for (int kb = 0; kb < 16; ++kb) {
    const int k0 = kb * 4 + 2 * lh;
    v2f a;                                   // A[m][k] = points[k][nbase+m]
    a.x = sb[(size_t)k0 * N + nbase + lm];
    a.y = sb[(size_t)(k0 + 1) * N + nbase + lm];
    #pragma unroll
    for (int t = 0; t < 4; ++t) {            // B[k][o] = W[o][k]
      v2f w = *(const v2f*)&W[(t * 16 + lm) * kC + k0];
      acc[t] = wmma4(a, w, acc[t]);
    }
  }
  float* db = dst + ((size_t)b * N + nbase) * kC;
  #pragma unroll
  for (int t = 0; t < 4; ++t)
    #pragma unroll
    for (int rr = 0; rr < 8; ++rr)
      db[(size_t)(rr + 8 * lh) * kC + t * 16 + lm] = acc[t][rr];
}

// ---------------------------------------------------------------------------
// Kernel 3: fused dual-KNN, async-LDS double-buffered.
// One wave owns 16 query rows; A-fragments (f1n rows + xyz) preloaded in
// VGPRs. Per 16-col tile: the NEXT tile's f2n block (4KB, contiguous) is
// DMA'd to LDS with global_load_async_to_lds_b128 while this tile runs its
// 16 cosine WMMAs + 1 xyz-dot WMMA. s_wait_asynccnt gates buffer reuse.
// Lanes 0-15 keep top-8 max-sim lists, lanes 16-31 top-8 min-dist lists.
// ---------------------------------------------------------------------------
__global__ __launch_bounds__(32) void knn_kernel(
    const float* __restrict__ f1n, const float* __restrict__ f2n,
    const float* __restrict__ x1t, const float* __restrict__ x2t,
    const float* __restrict__ n2sq, int* __restrict__ idx) {
  const int b = blockIdx.y;
  const int rbase = blockIdx.x * 16;
  const int l = threadIdx.x, lm = l & 15, lh = l >> 4;

  __shared__ __align__(16) float ftile[2][16][68];  // padded: stride 68 floats
  __shared__ float simt[16][17];
  __shared__ float sqt[16][17];
  __shared__ float n2s[16];

  // Preload A fragments (16 rows x 64 chans of f1n, + xyz) into VGPRs.
  v2f af[16];
  const float* f1p = &f1n[((size_t)b * kN1 + rbase + lm) * kC];
  #pragma unroll
  for (int kb = 0; kb < 16; ++kb) af[kb] = *(const v2f*)&f1p[kb * 4 + 2 * lh];
  const v2f ax = *(const v2f*)&x1t[((size_t)b * kN1 + rbase + lm) * 4 + 2 * lh];

  float bval[8];
  int   bidw[8];
  const bool ismin = (lh == 1);  // lanes 16-31: euclid(min); lanes 0-15: sim(max)
  #pragma unroll
  for (int i = 0; i < 8; ++i) { bval[i] = ismin ? 3.0e38f : -3.0e38f; bidw[i] = 0; }

  const float* f2b = &f2n[(size_t)b * kN2 * kC];
  const float* x2b = &x2t[(size_t)b * kN2 * 4];
  const float* n2b = &n2sq[(size_t)b * kN2];

  // Per-lane async-copy mapping: row = lm, half = lh -> 128 contiguous bytes.
  const unsigned laneByte = (unsigned)(lm * 256 + lh * 128);
  const unsigned ldsLane0 =
      (unsigned)(uintptr_t)&ftile[0][0][0] + (unsigned)(lm * 68 * 4 + lh * 128);
  const unsigned ldsBufStride = 16u * 68u * 4u;  // 4352 B

  // Prime buffer 0 with tile 0.
  issue_ftile_async(ldsLane0,
                    (unsigned long long)(uintptr_t)f2b + laneByte);

  constexpr int kTiles = kN2 / 16;
  for (int ct = 0; ct < kTiles; ++ct) {
    const int cbase = ct * 16;
    const int cur = ct & 1;

    if (ct + 2 < kTiles)  // L2 warm-up two tiles ahead
      __builtin_prefetch(f2b + (size_t)(cbase + 32) * kC + l * 32, 0, 1);

    if (ct + 1 < kTiles) {
      issue_ftile_async(ldsLane0 + (cur ^ 1) * ldsBufStride,
                        (unsigned long long)(uintptr_t)(f2b + (size_t)(cbase + 16) * kC) + laneByte);
      asm volatile("s_wait_asynccnt 0x8" ::: "memory");  // current buffer landed
    } else {
      asm volatile("s_wait_asynccnt 0x0" ::: "memory");
    }

    // 16 cosine WMMAs, B-fragments from LDS (bank-conflict-free via pad 68).
    const float* fp = &ftile[cur][lm][0];
    v8f acc = {};
    #pragma unroll
    for (int kb = 0; kb < 16; ++kb) {
      const int k0 = kb * 4 + 2 * lh;
      v2f bf;
      bf.x = fp[k0];
      bf.y = fp[k0 + 1];
      acc = wmma4(af[kb], bf, acc);
    }
    // xyz dot tile (K padded to 4), direct from global.
    v8f dac = {};
    {
      v2f bx = *(const v2f*)&x2b[((size_t)cbase + lm) * 4 + 2 * lh];
      dac = wmma4(ax, bx, dac);
    }
    if (l < 16) n2s[l] = n2b[cbase + l];
    #pragma unroll
    for (int rr = 0; rr < 8; ++rr) {
      simt[rr + 8 * lh][lm] = acc[rr];
      sqt[rr + 8 * lh][lm]  = dac[rr];
    }
    __syncthreads();

    const int row = lm;
    #pragma unroll
    for (int j = 0; j < 16; ++j) {
      const float v = ismin ? (n2s[j] - 2.0f * sqt[row][j]) : simt[row][j];
      const int id = cbase + j;
      const bool beatsWorst = ismin ? (v < bval[7]) : (v > bval[7]);
      if (beatsWorst) {
        float cv = v; int ci = id;
        #pragma unroll
        for (int s = 0; s < 8; ++s) {       // stable bubble-insert (strict cmp)
          const bool rep = ismin ? (cv < bval[s]) : (cv > bval[s]);
          if (rep) {
            float tv = bval[s]; int ti = bidw[s];
            bval[s] = cv; bidw[s] = ci; cv = tv; ci = ti;
          }
        }
      }
    }
    __syncthreads();
  }

  int* op = &idx[(((size_t)b * kN1) + rbase + lm) * kK + (ismin ? 8 : 0)];
  #pragma unroll
  for (int j = 0; j < 8; ++j) op[j] = bidw[j];
}

// ---------------------------------------------------------------------------
// Kernel 4 helpers: GRU gate layers over the 16 gathered neighbors.
// LDS activation layout: X[channel][neighbor], padded stride 17.
// ---------------------------------------------------------------------------
__device__ __forceinline__ void gate_layer0(
    float (*X)[17], const float (*Rl)[17], v2f bdir,
    const float* __restrict__ W0, const float* __restrict__ b0,
    const float* __restrict__ gatherB, const int* __restrict__ sidx,
    const float* __restrict__ pAdd, const float* __restrict__ rmul,
    int lm, int lh) {
  const int col = lm;
  const float* gp = gatherB + (size_t)sidx[col] * kC;  // gathered neighbor feats
  #pragma unroll
  for (int t = 0; t < 4; ++t) {
    const int oc = t * 16 + lm;
    v2f a;                                   // A[m][k] = W0[oc][k], K padded to 4
    a.x = W0[oc * 3 + 2 * lh];               // k = 0 or 2
    a.y = (lh == 0) ? W0[oc * 3 + 1] : 0.0f; // k = 1 or 3 (pad)
    v8f d = {};
    d = wmma4(a, bdir, d);
    #pragma unroll
    for (int rr = 0; rr < 8; ++rr) {
      const int c = t * 16 + rr + 8 * lh;
      float add = gp[c] + b0[c];
      if (pAdd) add += pAdd[c];
      if (rmul) add += Rl[c][col] * rmul[c]; // h gate: + r * p1o
      X[c][col] = leaky_(d[rr] + add);
    }
  }
}

__device__ __forceinline__ void gate_layer64(
    const float (*Xin)[17], float (*Xout)[17],
    const float* __restrict__ W1, const float* __restrict__ b1,
    int lm, int lh, bool sig) {
  v8f acc[4] = {};
  #pragma unroll
  for (int kb = 0; kb < 16; ++kb) {
    const int k0 = kb * 4 + 2 * lh;
    v2f bf;                                  // B[k][neighbor] from LDS
    bf.x = Xin[k0][lm];
    bf.y = Xin[k0 + 1][lm];
    #pragma unroll
    for (int t = 0; t < 4; ++t) {            // A[m][k] = W1[t*16+m][k]
      v2f a = *(const v2f*)&W1[(t * 16 + lm) * kC + k0];
      acc[t] = wmma4(a, bf, acc[t]);
    }
  }
  __syncthreads();                           // allow in-place Xout == Xin
  #pragma unroll
  for (int t = 0; t < 4; ++t)
    #pragma unroll
    for (int rr = 0; rr < 8; ++rr) {
      const int c = t * 16 + rr + 8 * lh;
      const float v = acc[t][rr] + b1[c];
      Xout[c][lm] = sig ? sigmoid_(v) : leaky_(v);
    }
}

__device__ __forceinline__ void colmax16(const float (*Xin)[17],
                                         float* __restrict__ vout, int l) {
  for (int c = l; c < kC; c += 32) {
    float m = Xin[c][0];
    #pragma unroll
    for (int k = 1; k < kK; ++k) m = fmaxf(m, Xin[c][k]);
    vout[c] = m;
  }
}

__device__ __forceinline__ void gemv64(const float* __restrict__ W,
                                       const float* __restrict__ bias,
                                       const float* __restrict__ vin,
                                       int l, float& o0, float& o1) {
  float a0 = bias[l], a1 = bias[l + 32];
  #pragma unroll 8
  for (int c = 0; c < kC; ++c) {
    const float m = vin[c];
    a0 += W[l * kC + c] * m;
    a1 += W[(l + 32) * kC + c] * m;
  }
  o0 = a0; o1 = a1;
}

// ---------------------------------------------------------------------------
// Kernel 4: per-point GRU (one wave per output point).
// ---------------------------------------------------------------------------
__global__ __launch_bounds__(32) void gru_kernel(
    const float* __restrict__ x1t, const float* __restrict__ x2t,
    const int* __restrict__ idx,
    const float* __restrict__ p2r, const float* __restrict__ p2ro,
    const float* __restrict__ p2z,
    const float* __restrict__ gp1r, const float* __restrict__ gp1z,
    const float* __restrict__ p1o,
    const float* __restrict__ Wr0, const float* __restrict__ br0,
    const float* __restrict__ Wr1, const float* __restrict__ br1,
    const float* __restrict__ Wr2, const float* __restrict__ br2,
    const float* __restrict__ Wz0, const float* __restrict__ bz0,
    const float* __restrict__ Wz1, const float* __restrict__ bz1,
    const float* __restrict__ Wz2, const float* __restrict__ bz2,
    const float* __restrict__ Wh0, const float* __restrict__ bh0,
    const float* __restrict__ Wh1, const float* __restrict__ bh1,
    const float* __restrict__ Wh2, const float* __restrict__ bh2,
    float* __restrict__ out) {
  const int n = blockIdx.x, b = blockIdx.y;
  const int l = threadIdx.x, lm = l & 15, lh = l >> 4;
  const size_t bn = (size_t)b * kN1 + n;

  __shared__ int   sidx[kK];
  __shared__ float X[kC][17];
  __shared__ float R[kC][17];
  __shared__ float vmax[kC];

  if (l < kK) sidx[l] = idx[bn * kK + l];
  __syncthreads();

  // dirx B-fragment: channel k = 2*lh + j, neighbor col = lm (pad chan == 0)
  v2f bdir;
  {
    const float* x2p = &x2t[((size_t)b * kN2 + sidx[lm]) * 4 + 2 * lh];
    const float* x1p = &x1t[bn * 4 + 2 * lh];
    bdir.x = x2p[0] - x1p[0];
    bdir.y = x2p[1] - x1p[1];
  }

  const float* p2r_b  = p2r  + (size_t)b * kN2 * kC;
  const float* p2ro_b = p2ro + (size_t)b * kN2 * kC;
  const float* p2z_b  = p2z  + (size_t)b * kN2 * kC;
  const float* p1o_n  = p1o  + bn * kC;

  // ---- r gate: leaky(W0·dirx + gp1r + g_r) -> leaky(W1·) -> sigmoid(W2·)
  gate_layer0(X, nullptr, bdir, Wr0, br0, p2r_b, sidx, gp1r + bn * kC, nullptr, lm, lh);
  __syncthreads();
  gate_layer64(X, X, Wr1, br1, lm, lh, false);
  __syncthreads();
  gate_layer64(X, R, Wr2, br2, lm, lh, true);
  __syncthreads();

  // ---- z gate: leaky(W0·dirx + gp1z + g_ro) -> leaky(W1·) -> max_k -> sigmoid(W2·)
  gate_layer0(X, nullptr, bdir, Wz0, bz0, p2ro_b, sidx, gp1z + bn * kC, nullptr, lm, lh);
  __syncthreads();
  gate_layer64(X, X, Wz1, bz1, lm, lh, false);
  __syncthreads();
  colmax16(X, vmax, l);
  __syncthreads();
  float z0, z1;
  gemv64(Wz2, bz2, vmax, l, z0, z1);
  z0 = sigmoid_(z0); z1 = sigmoid_(z1);
  __syncthreads();

  // ---- h gate: leaky(W0·dirx + r*p1o + g_z) -> leaky(W1·) -> max_k -> leaky(W2·)
  gate_layer0(X, R, bdir, Wh0, bh0, p2z_b, sidx, nullptr, p1o_n, lm, lh);
  __syncthreads();
  gate_layer64(X, X, Wh1, bh1, lm, lh, false);
  __syncthreads();
  colmax16(X, vmax, l);
  __syncthreads();
  float h0, h1;
  gemv64(Wh2, bh2, vmax, l, h0, h1);
  h0 = leaky_(h0); h1 = leaky_(h1);

  // ---- out = (1-z)*p1o + z*h, stored as [B, 64, N1]
  const float p0 = p1o_n[l], p1 = p1o_n[l + 32];
  out[((size_t)b * kC + l) * kN1 + n]      = (1.0f - z0) * p0 + z0 * h0;
  out[((size_t)b * kC + l + 32) * kN1 + n] = (1.0f - z1) * p1 + z1 * h1;
}

// ---------------------------------------------------------------------------
extern "C" void kernel_launch(void* const* d_in, const int* in_sizes, int n_in,
                              void* d_out, int out_size, void* d_ws, size_t ws_size,
                              hipStream_t stream) {
  const float* xyz1    = (const float*)d_in[0];
  const float* xyz2    = (const float*)d_in[1];
  const float* points1 = (const float*)d_in[2];
  const float* points2 = (const float*)d_in[3];
  const float* knn1    = (const float*)d_in[4];
  const float* knn2    = (const float*)d_in[5];
  const float* Wfr   = (const float*)d_in[6];
  const float* Wfro  = (const float*)d_in[7];
  const float* Wfz   = (const float*)d_in[8];
  const float* Wfr2  = (const float*)d_in[9];
  const float* Wfro2 = (const float*)d_in[10];
  const float* Wfz2  = (const float*)d_in[11];
  const float* Wr0 = (const float*)d_in[12]; const float* br0 = (const float*)d_in[13];
  const float* Wr1 = (const float*)d_in[14]; const float* br1 = (const float*)d_in[15];
  const float* Wr2 = (const float*)d_in[16]; const float* br2 = (const float*)d_in[17];
  const float* Wz0 = (const float*)d_in[18]; const float* bz0 = (const float*)d_in[19];
  const float* Wz1 = (const float*)d_in[20]; const float* bz1 = (const float*)d_in[21];
  const float* Wz2 = (const float*)d_in[22]; const float* bz2 = (const float*)d_in[23];
  const float* Wh0 = (const float*)d_in[24]; const float* bh0 = (const float*)d_in[25];
  const float* Wh1 = (const float*)d_in[26]; const float* bh1 = (const float*)d_in[27];
  const float* Wh2 = (const float*)d_in[28]; const float* bh2 = (const float*)d_in[29];
  float* out = (float*)d_out;

  // Workspace carve-up (floats).
  float* ws = (float*)d_ws;
  size_t off = 0;
  float* f1n  = ws + off; off += (size_t)kB * kN1 * kC;
  float* f2n  = ws + off; off += (size_t)kB * kN2 * kC;
  float* x1t  = ws + off; off += (size_t)kB * kN1 * 4;
  float* x2t  = ws + off; off += (size_t)kB * kN2 * 4;
  float* n2sq = ws + off; off += (size_t)kB * kN2;
  float* p2r  = ws + off; off += (size_t)kB * kN2 * kC;
  float* p2ro = ws + off; off += (size_t)kB * kN2 * kC;
  float* p2z  = ws + off; off += (size_t)kB * kN2 * kC;
  float* gp1r = ws + off; off += (size_t)kB * kN1 * kC;
  float* gp1z = ws + off; off += (size_t)kB * kN1 * kC;
  float* p1o  = ws + off; off += (size_t)kB * kN1 * kC;
  int*   nidx = (int*)(ws + off); off += (size_t)kB * kN1 * kK;

  prep_kernel<<<dim3(kN2 / 256, 2, kB), 256, 0, stream>>>(
      xyz1, xyz2, knn1, knn2, f1n, f2n, x1t, x2t, n2sq);

  fuse_kernel<<<dim3(kN2 / 16, 6, kB), 32, 0, stream>>>(
      points1, points2, Wfr, Wfro, Wfz, Wfr2, Wfro2, Wfz2,
      gp1r, p1o, gp1z, p2r, p2ro, p2z);

  knn_kernel<<<dim3(kN1 / 16, kB), 32, 0, stream>>>(
      f1n, f2n, x1t, x2t, n2sq, nidx);

  gru_kernel<<<dim3(kN1, kB), 32, 0, stream>>>(
      x1t, x2t, nidx, p2r, p2ro, p2z, gp1r, gp1z, p1o,
      Wr0, br0, Wr1, br1, Wr2, br2,
      Wz0, bz0, Wz1, bz1, Wz2, bz2,
      Wh0, bh0, Wh1, bh1, Wh2, bh2, out);
}